// NeuralNetwork_54503134986667
// MI455X (gfx1250) — compile-verified
//
#include <hip/hip_runtime.h>
#include <math.h>

typedef __attribute__((ext_vector_type(2))) float v2f;
typedef __attribute__((ext_vector_type(8))) float v8f;

#define N_NODES 100000
#define N_EDGES 1000000

// ---------------------------------------------------------------------------
// zero-init workspace region
// ---------------------------------------------------------------------------
__global__ void zero_kernel(float* __restrict__ p, int n) {
    int i = blockIdx.x * blockDim.x + threadIdx.x;
    if (i < n) p[i] = 0.0f;
}

// ---------------------------------------------------------------------------
// Layer 1 edge kernel.
// Per wave: 16 edges. Edge MLP [16,4] @ [4,192(pad of 180)] via 12x
// V_WMMA_F32_16X16X4_F32 -> per-edge weights (18x10) staged in LDS.
// Then 2 threads/edge compute msg = x[src] (18) * w (18x10) and atomic-add
// into agg1[dst].
// ---------------------------------------------------------------------------
__global__ void __launch_bounds__(128) edge1_kernel(
    const float* __restrict__ x, const long long* __restrict__ edge_index,
    const float* __restrict__ edge_attr, const float* __restrict__ W1e,
    const float* __restrict__ b1e, float* __restrict__ agg1)
{
    __shared__ float w_lds[64 * 193];   // 64 edges x 180 (stride 193 vs bank conflicts)

    const int lane = threadIdx.x & 31;
    const int wave = threadIdx.x >> 5;
    const int m    = lane & 15;         // tile row (edge) / B column within tile
    const int kh   = lane >> 4;         // K half: 0 -> K{0,1}, 1 -> K{2,3}
    const long long eBase = (long long)blockIdx.x * 64;
    const long long e0    = eBase + (long long)wave * 16;

    // A operand: edge_attr tile, per-lane float2
    const float* ap = edge_attr + (e0 + m) * 4 + kh * 2;
    v2f a;
    a.x = ap[0];
    a.y = ap[1];

    #pragma unroll
    for (int t = 0; t < 12; ++t) {
        const int n = t * 16 + m;       // output-feature column of this lane
        v2f b;
        if (n < 180) {
            b.x = W1e[n * 4 + kh * 2];
            b.y = W1e[n * 4 + kh * 2 + 1];
        } else {
            b.x = 0.0f; b.y = 0.0f;
        }
        v8f c = {};
        c = __builtin_amdgcn_wmma_f32_16x16x4_f32(
                /*neg_a=*/false, a, /*neg_b=*/false, b,
                /*c_mod=*/(short)0, c, /*reuse_a=*/false, /*reuse_b=*/false);
        if (n < 180) {
            const float bias = b1e[n];
            #pragma unroll
            for (int r = 0; r < 8; ++r) {
                const int eloc = wave * 16 + kh * 8 + r;   // D row = r + kh*8
                w_lds[eloc * 193 + n] = c[r] + bias;
            }
        }
    }
    __syncthreads();

    // Message phase: 128 threads, 64 edges -> 2 threads per edge (5 outs each)
    const int eloc = threadIdx.x >> 1;
    const int half = threadIdx.x & 1;
    const long long e   = eBase + eloc;
    const long long src = edge_index[e];
    const long long dst = edge_index[N_EDGES + e];
    const float* xs = x + src * 18;
    const float* wl = &w_lds[eloc * 193];
    float acc[5] = {0.f, 0.f, 0.f, 0.f, 0.f};
    #pragma unroll
    for (int i = 0; i < 18; ++i) {
        const float xv = xs[i];
        #pragma unroll
        for (int o = 0; o < 5; ++o)
            acc[o] += xv * wl[i * 10 + half * 5 + o];
    }
    float* dp = agg1 + dst * 10 + half * 5;
    #pragma unroll
    for (int o = 0; o < 5; ++o) atomicAdd(dp + o, acc[o]);
}

// ---------------------------------------------------------------------------
// h1 = agg1 + x @ Wr1 + bb1   (one thread per (node, out-channel))
// ---------------------------------------------------------------------------
__global__ void h1_kernel(const float* __restrict__ x, const float* __restrict__ agg1,
                          const float* __restrict__ Wr1, const float* __restrict__ bb1,
                          float* __restrict__ h1)
{
    int idx = blockIdx.x * blockDim.x + threadIdx.x;
    if (idx >= N_NODES * 10) return;
    const int n = idx / 10;
    const int o = idx - n * 10;
    float s = agg1[idx] + bb1[o];
    const float* xs = x + (long long)n * 18;
    #pragma unroll
    for (int i = 0; i < 18; ++i) s += xs[i] * Wr1[i * 10 + o];
    h1[idx] = s;
}

// ---------------------------------------------------------------------------
// Layer 2 edge kernel: edge MLP [16,4] @ [4,80(pad of 70)] via 5x WMMA,
// per-edge weights (10x7) in LDS, msg = h1[src] * w, atomic-add into agg2.
// ---------------------------------------------------------------------------
__global__ void __launch_bounds__(128) edge2_kernel(
    const float* __restrict__ h1, const long long* __restrict__ edge_index,
    const float* __restrict__ edge_attr, const float* __restrict__ W2e,
    const float* __restrict__ b2e, float* __restrict__ agg2)
{
    __shared__ float w_lds[64 * 81];

    const int lane = threadIdx.x & 31;
    const int wave = threadIdx.x >> 5;
    const int m    = lane & 15;
    const int kh   = lane >> 4;
    const long long eBase = (long long)blockIdx.x * 64;
    const long long e0    = eBase + (long long)wave * 16;

    const float* ap = edge_attr + (e0 + m) * 4 + kh * 2;
    v2f a;
    a.x = ap[0];
    a.y = ap[1];

    #pragma unroll
    for (int t = 0; t < 5; ++t) {
        const int n = t * 16 + m;
        v2f b;
        if (n < 70) {
            b.x = W2e[n * 4 + kh * 2];
            b.y = W2e[n * 4 + kh * 2 + 1];
        } else {
            b.x = 0.0f; b.y = 0.0f;
        }
        v8f c = {};
        c = __builtin_amdgcn_wmma_f32_16x16x4_f32(
                false, a, false, b, (short)0, c, false, false);
        if (n < 70) {
            const float bias = b2e[n];
            #pragma unroll
            for (int r = 0; r < 8; ++r) {
                const int eloc = wave * 16 + kh * 8 + r;
                w_lds[eloc * 81 + n] = c[r] + bias;
            }
        }
    }
    __syncthreads();

    const int eloc = threadIdx.x >> 1;
    const int half = threadIdx.x & 1;   // half 0 -> outs 0..3, half 1 -> outs 4..6
    const long long e   = eBase + eloc;
    const long long src = edge_index[e];
    const long long dst = edge_index[N_EDGES + e];
    const float* hs = h1 + src * 10;
    const float* wl = &w_lds[eloc * 81];
    const int o0 = half * 4;
    float acc[4] = {0.f, 0.f, 0.f, 0.f};
    #pragma unroll
    for (int i = 0; i < 10; ++i) {
        const float hv = hs[i];
        #pragma unroll
        for (int o = 0; o < 4; ++o)
            if (o0 + o < 7) acc[o] += hv * wl[i * 7 + o0 + o];
    }
    #pragma unroll
    for (int o = 0; o < 4; ++o)
        if (o0 + o < 7) atomicAdd(&agg2[dst * 7 + o0 + o], acc[o]);
}

// ---------------------------------------------------------------------------
// h2 = agg2 + h1 @ Wr2 + bb2, fused with mean-pool partial reduction
// ---------------------------------------------------------------------------
__global__ void h2_pool_kernel(const float* __restrict__ h1, const float* __restrict__ agg2,
                               const float* __restrict__ Wr2, const float* __restrict__ bb2,
                               float* __restrict__ gsum)
{
    __shared__ float ssum[7];
    if (threadIdx.x < 7) ssum[threadIdx.x] = 0.0f;
    __syncthreads();

    const int n = blockIdx.x * blockDim.x + threadIdx.x;
    if (n < N_NODES) {
        float hv[10];
        #pragma unroll
        for (int i = 0; i < 10; ++i) hv[i] = h1[(long long)n * 10 + i];
        #pragma unroll
        for (int o = 0; o < 7; ++o) {
            float s = agg2[(long long)n * 7 + o] + bb2[o];
            #pragma unroll
            for (int i = 0; i < 10; ++i) s += hv[i] * Wr2[i * 7 + o];
            atomicAdd(&ssum[o], s);
        }
    }
    __syncthreads();
    if (threadIdx.x < 7) atomicAdd(&gsum[threadIdx.x], ssum[threadIdx.x]);
}

// ---------------------------------------------------------------------------
// Final MLP: mean -> 7 -> 20 -> 10 -> 1 -> sigmoid (single thread)
// ---------------------------------------------------------------------------
__global__ void final_kernel(const float* __restrict__ gsum,
                             const float* __restrict__ w1, const float* __restrict__ bw1,
                             const float* __restrict__ w2, const float* __restrict__ bw2,
                             const float* __restrict__ w3, const float* __restrict__ bw3,
                             float* __restrict__ out)
{
    if (threadIdx.x != 0 || blockIdx.x != 0) return;
    float g[7];
    #pragma unroll
    for (int o = 0; o < 7; ++o) g[o] = gsum[o] * (1.0f / (float)N_NODES);
    float t1[20];
    for (int j = 0; j < 20; ++j) {
        float s = bw1[j];
        #pragma unroll
        for (int o = 0; o < 7; ++o) s += g[o] * w1[j * 7 + o];
        t1[j] = s > 0.0f ? s : 0.0f;
    }
    float t2[10];
    for (int k = 0; k < 10; ++k) {
        float s = bw2[k];
        for (int j = 0; j < 20; ++j) s += t1[j] * w2[k * 20 + j];
        t2[k] = s > 0.0f ? s : 0.0f;
    }
    float s = bw3[0];
    #pragma unroll
    for (int k = 0; k < 10; ++k) s += t2[k] * w3[k];
    out[0] = 1.0f / (1.0f + expf(-s));
}

// ---------------------------------------------------------------------------
extern "C" void kernel_launch(void* const* d_in, const int* in_sizes, int n_in,
                              void* d_out, int out_size, void* d_ws, size_t ws_size,
                              hipStream_t stream)
{
    (void)in_sizes; (void)n_in; (void)out_size; (void)ws_size;

    const float*     x    = (const float*)d_in[0];
    const long long* ei   = (const long long*)d_in[1];   // int64 per reference
    const float*     ea   = (const float*)d_in[2];
    const float*     W1e  = (const float*)d_in[3];
    const float*     b1e  = (const float*)d_in[4];
    const float*     Wr1  = (const float*)d_in[5];
    const float*     bb1  = (const float*)d_in[6];
    const float*     W2e  = (const float*)d_in[7];
    const float*     b2e  = (const float*)d_in[8];
    const float*     Wr2  = (const float*)d_in[9];
    const float*     bb2  = (const float*)d_in[10];
    const float*     w1   = (const float*)d_in[11];
    const float*     bw1  = (const float*)d_in[12];
    const float*     w2   = (const float*)d_in[13];
    const float*     bw2  = (const float*)d_in[14];
    const float*     w3   = (const float*)d_in[15];
    const float*     bw3  = (const float*)d_in[16];
    float* out = (float*)d_out;

    // workspace layout (floats): agg1[1e6] | agg2[7e5] | gsum[8] | h1[1e6]
    float* ws   = (float*)d_ws;
    float* agg1 = ws;
    float* agg2 = agg1 + 1000000;
    float* gsum = agg2 + 700000;
    float* h1   = gsum + 8;

    const int zn = 1000000 + 700000 + 8;
    zero_kernel<<<(zn + 255) / 256, 256, 0, stream>>>(ws, zn);

    // 64 edges per block, 1M / 64 = 15625 blocks exactly
    edge1_kernel<<<15625, 128, 0, stream>>>(x, ei, ea, W1e, b1e, agg1);
    h1_kernel<<<(N_NODES * 10 + 255) / 256, 256, 0, stream>>>(x, agg1, Wr1, bb1, h1);
    edge2_kernel<<<15625, 128, 0, stream>>>(h1, ei, ea, W2e, b2e, agg2);
    h2_pool_kernel<<<(N_NODES + 255) / 256, 256, 0, stream>>>(h1, agg2, Wr2, bb2, gsum);
    final_kernel<<<1, 32, 0, stream>>>(gsum, w1, bw1, w2, bw2, w3, bw3, out);
}